// Dense_34643206210174
// MI455X (gfx1250) — compile-verified
//
#include <hip/hip_runtime.h>
#include <stdint.h>

typedef __attribute__((ext_vector_type(2))) float v2f;
typedef __attribute__((ext_vector_type(8))) float v8f;

#define M_TOT 8192
#define K_TOT 4096
#define N_TOT 16384
#define BM 128
#define BN 128
#define BK 16
#define LDA 20    // padded A-tile row stride (floats): 80B, 16B-aligned rows, conflict-free b64 frag reads
#define LDB 144   // padded B-tile row stride (floats): 576B, half-waves hit disjoint bank groups
#define DEPTH 4   // async pipeline stages (power of two)

#define A_TILE (BM * LDA)   // 2560 floats = 10240 B (16B multiple)
#define B_TILE (BK * LDB)   // 2304 floats =  9216 B (16B multiple)

// Low 32 bits of a flat LDS address are the LDS byte offset (aperture scheme).
__device__ __forceinline__ uint32_t lds_off(const void* p) {
    return (uint32_t)(uintptr_t)p;
}

// CDNA5 async copy: per-lane 16B global -> LDS, tracked by ASYNCcnt, no VGPR data.
__device__ __forceinline__ void async_copy_b128(const float* gptr, uint32_t lds_byte_off) {
    asm volatile("global_load_async_to_lds_b128 %0, %1, off"
                 :: "v"(lds_byte_off), "v"(gptr)
                 : "memory");
}

__launch_bounds__(256)
__global__ void dense_wmma_f32_async(const float* __restrict__ A,
                                     const float* __restrict__ B,
                                     const float* __restrict__ bias,
                                     float* __restrict__ C) {
    __shared__ __align__(16) float Alds[DEPTH][A_TILE];
    __shared__ __align__(16) float Blds[DEPTH][B_TILE];

    const int tid  = threadIdx.x;
    const int lane = tid & 31;
    const int wave = tid >> 5;     // 0..7
    const int wr   = wave >> 1;    // 0..3 : 32 output rows per wave
    const int wc   = wave & 1;     // 0..1 : 64 output cols per wave

    // L2 supertile swizzle: panels of 8 N-tiles; walk all M inside a panel.
    // Per panel: B panel (16 MB) + all of A (128 MB) <= 144 MB < 192 MB L2.
    const int numBM = M_TOT / BM;      // 64
    const int GROUP = 8;
    const int bid   = blockIdx.x;
    const int panel = bid / (numBM * GROUP);
    const int inp   = bid % (numBM * GROUP);
    const int bn    = panel * GROUP + (inp % GROUP);   // 0..127
    const int bm    = inp / GROUP;                     // 0..63

    // staging assignment: 256 threads, 4 x b128 async copies each per K-chunk
    const int rowA = tid >> 2;         // 0..63 (+64 second pass)
    const int colA = (tid & 3) * 4;    // 0,4,8,12
    const int rowB = tid >> 5;         // 0..7  (+8 second pass)
    const int colB = (tid & 31) * 4;   // 0..124

    const float* Ag = A + (size_t)(bm * BM) * K_TOT;
    const float* Bg = B + (size_t)(bn * BN);

    const int nK = K_TOT / BK;         // 256

    auto issue_stage = [&](int kt) {
        int k0 = kt * BK;
        if (k0 > K_TOT - BK) k0 = K_TOT - BK;   // tail dummies: keep ASYNCcnt uniform
        const int buf = kt & (DEPTH - 1);
        const float* ap = Ag + (size_t)rowA * K_TOT + k0 + colA;
        async_copy_b128(ap,                    lds_off(&Alds[buf][rowA * LDA + colA]));
        async_copy_b128(ap + (size_t)64 * K_TOT, lds_off(&Alds[buf][(rowA + 64) * LDA + colA]));
        const float* bp = Bg + (size_t)(k0 + rowB) * N_TOT + colB;
        async_copy_b128(bp,                    lds_off(&Blds[buf][rowB * LDB + colB]));
        async_copy_b128(bp + (size_t)8 * N_TOT, lds_off(&Blds[buf][(rowB + 8) * LDB + colB]));
    };

    // prologue: fill the pipeline
#pragma unroll
    for (int s = 0; s < DEPTH; ++s) issue_stage(s);

    v8f acc[2][4];
#pragma unroll
    for (int mt = 0; mt < 2; ++mt)
#pragma unroll
        for (int nt = 0; nt < 4; ++nt)
            acc[mt][nt] = (v8f){0.f, 0.f, 0.f, 0.f, 0.f, 0.f, 0.f, 0.f};

    const int mlane = lane & 15;
    const int khalf = lane >> 4;   // 0 or 1

    for (int kt = 0; kt < nK; ++kt) {
        // async loads complete in order: <=12 outstanding  =>  stage kt fully resident
        asm volatile("s_wait_asynccnt 12" ::: "memory");
        __syncthreads();           // all threads' stage-kt portions now visible

        const int buf = kt & (DEPTH - 1);
        const float* Abuf = &Alds[buf][0];
        const float* Bbuf = &Blds[buf][0];

#pragma unroll
        for (int kk = 0; kk < 4; ++kk) {
            // A 16x4 f32 frag: lane m = L%16; VGPR0/1 = K = 2*(L/16), 2*(L/16)+1
            const int kb = kk * 4 + 2 * khalf;
            v2f afrag[2];
#pragma unroll
            for (int mt = 0; mt < 2; ++mt) {
                const int row = wr * 32 + mt * 16 + mlane;
                afrag[mt] = *(const v2f*)&Abuf[row * LDA + kb];
            }
#pragma unroll
            for (int nt = 0; nt < 4; ++nt) {
                const int col = wc * 64 + nt * 16 + mlane;
                v2f bfrag;
                bfrag.x = Bbuf[kb * LDB + col];
                bfrag.y = Bbuf[(kb + 1) * LDB + col];
#pragma unroll
                for (int mt = 0; mt < 2; ++mt) {
                    acc[mt][nt] = __builtin_amdgcn_wmma_f32_16x16x4_f32(
                        /*neg_a=*/false, afrag[mt],
                        /*neg_b=*/false, bfrag,
                        /*c_mod=*/(short)0, acc[mt][nt],
                        /*reuse_a=*/false, /*reuse_b=*/false);
                }
            }
        }
        __syncthreads();           // buffer fully consumed by every wave
        issue_stage(kt + DEPTH);   // refill the just-freed buffer
    }

    // epilogue: bias add + store. C/D layout: VGPR r holds M = r + 8*(lane/16), N = lane%16.
#pragma unroll
    for (int nt = 0; nt < 4; ++nt) {
        const int gcol = bn * BN + wc * 64 + nt * 16 + mlane;
        const float bv = bias[gcol];
#pragma unroll
        for (int mt = 0; mt < 2; ++mt) {
            const int rbase = bm * BM + wr * 32 + mt * 16 + 8 * khalf;
#pragma unroll
            for (int r = 0; r < 8; ++r) {
                C[(size_t)(rbase + r) * N_TOT + gcol] = acc[mt][nt][r] + bv;
            }
        }
    }
}

extern "C" void kernel_launch(void* const* d_in, const int* in_sizes, int n_in,
                              void* d_out, int out_size, void* d_ws, size_t ws_size,
                              hipStream_t stream) {
    (void)in_sizes; (void)n_in; (void)out_size; (void)d_ws; (void)ws_size;
    const float* A    = (const float*)d_in[0];   // (4,2048,4096) -> (8192,4096)
    const float* B    = (const float*)d_in[1];   // (4096,16384)
    const float* bias = (const float*)d_in[2];   // (16384,)
    float* out        = (float*)d_out;           // (8192,16384)

    dim3 grid((M_TOT / BM) * (N_TOT / BN));      // 64 * 128 = 8192 blocks
    dim3 block(256);
    dense_wmma_f32_async<<<grid, block, 0, stream>>>(A, B, bias, out);
}